// GE_FixPoint_PredNet_54168127537417
// MI455X (gfx1250) — compile-verified
//
#include <hip/hip_runtime.h>
#include <math.h>

// ---------------------------------------------------------------------------
// GCN pipeline for MI455X (gfx1250, wave32, WMMA).
//   deg -> dis = rsqrt(deg+2)
//   h   = relu(x @ W_in + b_in)                       (WMMA f32 16x16x4)
//   2x: hw = h @ W                                    (WMMA f32 16x16x4)
//       agg = 2*dis^2*hw  (self loop, improved=True)
//       agg[col] += dis[row]*dis[col]*hw[row]         (edge scatter, f32 atomics)
//       h = relu(agg + b)
//   out = h @ W_out + b_out
// Feature rows padded to 16 floats so WMMA A-fragments load unguarded and
// edge gathers are float4; the 20MB feature arrays live in the 192MB L2.
// B panel staged zero-padded in LDS so the WMMA inner loop is branch-free.
// ---------------------------------------------------------------------------

#define LAT  10
#define LPAD 16
#define KMAX 128   // max padded K (input GEMM)

typedef float v2f __attribute__((ext_vector_type(2)));
typedef float v8f __attribute__((ext_vector_type(8)));

__global__ void k_init_cnt(int* __restrict__ cnt, int n) {
    int i = blockIdx.x * blockDim.x + threadIdx.x;
    if (i < n) cnt[i] = 0;
}

__global__ void k_count(const int* __restrict__ col, int* __restrict__ cnt, int e) {
    int i = blockIdx.x * blockDim.x + threadIdx.x;
    if (i < e) atomicAdd(&cnt[col[i]], 1);
}

__global__ void k_dis(const int* __restrict__ cnt, float* __restrict__ dis, int n) {
    int i = blockIdx.x * blockDim.x + threadIdx.x;
    if (i < n) dis[i] = rsqrtf((float)cnt[i] + 2.0f);
}

// C[M,16] = op(A[M,lda] @ B[kact,LAT])  tiled 16x16 per wave via WMMA f32 16x16x4.
// A must have >= ksteps*4 readable columns per row (zeros past kact).
// B is staged zero-padded into LDS so the inner loop has no divergent guards.
__global__ void k_gemm_wmma(const float* __restrict__ A, int lda,
                            const float* __restrict__ B, int kact, int ksteps,
                            const float* __restrict__ bias, int do_relu,
                            float* __restrict__ C) {
    __shared__ float Bs[KMAX * 16];   // [k][n], zero padded
    __shared__ float sBias[16];       // zero padded

    int kpad = ksteps * 4;
    for (int idx = threadIdx.x; idx < kpad * 16; idx += blockDim.x) {
        int k = idx >> 4;
        int n = idx & 15;
        Bs[idx] = (k < kact && n < LAT) ? B[k * LAT + n] : 0.0f;
    }
    if (threadIdx.x < 16) {
        sBias[threadIdx.x] =
            (bias != nullptr && threadIdx.x < LAT) ? bias[threadIdx.x] : 0.0f;
    }
    __syncthreads();

    int wave = (int)((blockIdx.x * blockDim.x + threadIdx.x) >> 5);
    int lane = threadIdx.x & 31;
    int half = lane >> 4;     // 0: lanes 0-15, 1: lanes 16-31
    int r    = lane & 15;     // A: row within tile; B/C: column
    long baseM = (long)wave * 16;
    const float* __restrict__ Arow = A + (baseM + r) * (long)lda;

    v8f c = {0.f, 0.f, 0.f, 0.f, 0.f, 0.f, 0.f, 0.f};
#pragma unroll 4
    for (int ks = 0; ks < ksteps; ++ks) {
        int ka = ks * 4 + half * 2;   // K pair held by this lane (A and B layouts match)
        v2f a, b;
        a.x = Arow[ka];
        a.y = Arow[ka + 1];
        b.x = Bs[ka * 16 + r];
        b.y = Bs[ka * 16 + 16 + r];
        c = __builtin_amdgcn_wmma_f32_16x16x4_f32(
                /*neg_a=*/false, a, /*neg_b=*/false, b,
                /*c_mod=*/(short)0, c, /*reuse_a=*/false, /*reuse_b=*/false);
    }

#pragma unroll
    for (int j = 0; j < 8; ++j) {
        int m = (int)baseM + j + half * 8;  // C/D layout: VGPR j -> M=j (+8 for hi lanes)
        int n = r;
        float v = c[j] + sBias[n];
        if (do_relu) v = fmaxf(v, 0.0f);
        C[(long)m * LPAD + n] = (n < LAT) ? v : 0.0f;
    }
}

// agg = 2 * dis^2 * hw   (self-loop of improved GCN); pad cols stay zero
__global__ void k_init_agg(const float* __restrict__ dis, const float* __restrict__ hw,
                           float* __restrict__ agg, int n16) {
    int i = blockIdx.x * blockDim.x + threadIdx.x;
    if (i >= n16) return;
    float d = dis[i >> 4];
    agg[i] = 2.0f * d * d * hw[i];
}

// agg[col] += dis[row]*dis[col] * hw[row]
__global__ void k_scatter(const int* __restrict__ row, const int* __restrict__ col,
                          const float* __restrict__ dis, const float* __restrict__ hw,
                          float* __restrict__ agg, int e) {
    int i = blockIdx.x * blockDim.x + threadIdx.x;
    if (i >= e) return;
    int rr = row[i];
    int cc = col[i];
    float nrm = dis[rr] * dis[cc];
    const float4* __restrict__ hp = (const float4*)(hw + (long)rr * LPAD);
    float4 h0 = hp[0];
    float4 h1 = hp[1];
    float4 h2 = hp[2];            // .x,.y valid (cols 8,9); .z,.w are pad zeros
    float* ap = agg + (long)cc * LPAD;
    atomicAdd(ap + 0, nrm * h0.x);
    atomicAdd(ap + 1, nrm * h0.y);
    atomicAdd(ap + 2, nrm * h0.z);
    atomicAdd(ap + 3, nrm * h0.w);
    atomicAdd(ap + 4, nrm * h1.x);
    atomicAdd(ap + 5, nrm * h1.y);
    atomicAdd(ap + 6, nrm * h1.z);
    atomicAdd(ap + 7, nrm * h1.w);
    atomicAdd(ap + 8, nrm * h2.x);
    atomicAdd(ap + 9, nrm * h2.y);
}

// h = relu(agg + b), keep pad cols zero
__global__ void k_combine(const float* __restrict__ agg, const float* __restrict__ bias,
                          float* __restrict__ h, int n16) {
    int i = blockIdx.x * blockDim.x + threadIdx.x;
    if (i >= n16) return;
    int j = i & 15;
    float v = 0.0f;
    if (j < LAT) v = fmaxf(agg[i] + bias[j], 0.0f);
    h[i] = v;
}

// out[i] = h[i,:] . W_out + b_out
__global__ void k_out(const float* __restrict__ h, const float* __restrict__ w,
                      const float* __restrict__ b, float* __restrict__ out, int n) {
    int i = blockIdx.x * blockDim.x + threadIdx.x;
    if (i >= n) return;
    const float* hr = h + (long)i * LPAD;
    float s = b[0];
#pragma unroll
    for (int j = 0; j < LAT; ++j) s += hr[j] * w[j];
    out[i] = s;
}

extern "C" void kernel_launch(void* const* d_in, const int* in_sizes, int n_in,
                              void* d_out, int out_size, void* d_ws, size_t ws_size,
                              hipStream_t stream) {
    const float* x     = (const float*)d_in[0];
    const int*   ei    = (const int*)d_in[1];
    const float* W_in  = (const float*)d_in[2];
    const float* b_in  = (const float*)d_in[3];
    const float* W1    = (const float*)d_in[4];
    const float* b1    = (const float*)d_in[5];
    const float* W2    = (const float*)d_in[6];
    const float* b2    = (const float*)d_in[7];
    const float* W_out = (const float*)d_in[8];
    const float* b_out = (const float*)d_in[9];
    float* out = (float*)d_out;

    const int F = 128;
    const int N = in_sizes[0] / F;        // 320000
    const int E = in_sizes[1] / 2;        // 10240000
    const int* row = ei;
    const int* col = ei + E;

    // Workspace layout (all offsets 256B aligned)
    char* ws = (char*)d_ws;
    size_t nb   = (size_t)N * 4;          // 1,280,000
    size_t nb16 = (size_t)N * LPAD * 4;   // 20,480,000
    int*   cnt = (int*)  (ws);
    float* dis = (float*)(ws + nb);
    float* h   = (float*)(ws + 2 * nb);
    float* hw  = (float*)(ws + 2 * nb + nb16);
    float* agg = (float*)(ws + 2 * nb + 2 * nb16);

    const int T = 256;
    int gN   = (N + T - 1) / T;
    int gE   = (E + T - 1) / T;
    int gN16 = (N * LPAD + T - 1) / T;
    int tiles = N / 16;                    // 20000 (N divisible by 16)
    int gGemm = tiles / 4;                 // 4 waves (16-row tiles) per 128-thread block

    // degree / normalization
    k_init_cnt<<<gN, T, 0, stream>>>(cnt, N);
    k_count<<<gE, T, 0, stream>>>(col, cnt, E);
    k_dis<<<gN, T, 0, stream>>>(cnt, dis, N);

    // h = relu(x @ W_in + b_in)   K=128 -> 32 WMMA steps
    k_gemm_wmma<<<gGemm, 128, 0, stream>>>(x, F, W_in, F, F / 4, b_in, 1, h);

    // layer 1
    k_gemm_wmma<<<gGemm, 128, 0, stream>>>(h, LPAD, W1, LAT, 3, nullptr, 0, hw);
    k_init_agg<<<gN16, T, 0, stream>>>(dis, hw, agg, N * LPAD);
    k_scatter<<<gE, T, 0, stream>>>(row, col, dis, hw, agg, E);
    k_combine<<<gN16, T, 0, stream>>>(agg, b1, h, N * LPAD);

    // layer 2
    k_gemm_wmma<<<gGemm, 128, 0, stream>>>(h, LPAD, W2, LAT, 3, nullptr, 0, hw);
    k_init_agg<<<gN16, T, 0, stream>>>(dis, hw, agg, N * LPAD);
    k_scatter<<<gE, T, 0, stream>>>(row, col, dis, hw, agg, E);
    k_combine<<<gN16, T, 0, stream>>>(agg, b2, h, N * LPAD);

    // out = h @ W_out + b_out
    k_out<<<gN, T, 0, stream>>>(h, W_out, b_out, out, N);
}